// GNNEncoder_1855425872439
// MI455X (gfx1250) — compile-verified
//
#include <hip/hip_runtime.h>
#include <hip/hip_bf16.h>

typedef __attribute__((ext_vector_type(2))) float v2f;
typedef __attribute__((ext_vector_type(8))) float v8f;
typedef __attribute__((ext_vector_type(4))) unsigned int v4u;
typedef __attribute__((ext_vector_type(8))) int v8i;
typedef __attribute__((ext_vector_type(4))) int v4i;

#define IN_C  128
#define HID_C 128
#define OUT_C 64

// ---------------------------------------------------------------------------
// Degree / normalization kernels
// ---------------------------------------------------------------------------
__global__ void deg_init_kernel(float* __restrict__ deg, int n) {
  int i = blockIdx.x * blockDim.x + threadIdx.x;
  if (i < n) deg[i] = 1.0f;  // self-loop contributes 1 to every node's degree
}

__global__ void deg_accum_kernel(const int* __restrict__ dst, float* __restrict__ deg, int e) {
  int i = blockIdx.x * blockDim.x + threadIdx.x;
  if (i < e) atomicAdd(&deg[dst[i]], 1.0f);
}

__global__ void dinv_kernel(const float* __restrict__ deg, float* __restrict__ dinv, int n) {
  int i = blockIdx.x * blockDim.x + threadIdx.x;
  if (i < n) dinv[i] = rsqrtf(deg[i]);  // deg >= 1 always (self-loop)
}

// ---------------------------------------------------------------------------
// TDM: DMA the whole weight matrix (K x N f32, row-major) into LDS.
// D# descriptor per CDNA5 ISA ch.8: group0 = {count/type/lds_addr/global_addr},
// group1 = {data_size=4B, tensor dims, tile dims, stride}, groups 2/3 zero.
// Issued by one wave; completion via S_WAIT_TENSORCNT.
// 6-arg builtin form (clang-23 / therock-10.0 flavor).
// ---------------------------------------------------------------------------
__device__ __forceinline__ void tdm_load_w(const float* W, unsigned ldsOff,
                                           int nDim, int kDim) {
  const unsigned long long ga = (unsigned long long)(uintptr_t)W;
  v4u g0;
  g0.x = 1u;                                            // count=1 (valid user D#)
  g0.y = ldsOff;                                        // lds_addr (bytes)
  g0.z = (unsigned)(ga & 0xFFFFFFFFull);                // global_addr[31:0]
  g0.w = (unsigned)((ga >> 32) & 0x01FFFFFFull)         // global_addr[56:32]
         | (2u << 30);                                  // type = 2 ("image")
  v8i g1;
  g1[0] = (2 << 16);                                    // wg_mask=0, data_size=4B
  g1[1] = (nDim & 0xFFFF) << 16;                        // tensor_dim0 lo -> [63:48]
  g1[2] = ((nDim >> 16) & 0xFFFF) | ((kDim & 0xFFFF) << 16);  // tdim0 hi | tdim1 lo
  g1[3] = ((kDim >> 16) & 0xFFFF) | ((nDim & 0xFFFF) << 16);  // tdim1 hi | tile_dim0
  g1[4] = (kDim & 0xFFFF);                              // tile_dim1=K, tile_dim2=0
  g1[5] = nDim;                                         // tensor_dim0_stride lo32
  g1[6] = 0;                                            // stride0 hi | stride1 lo
  g1[7] = 0;                                            // stride1 hi (2-D: unused)
  v4i g2 = {0, 0, 0, 0};
  v4i g3 = {0, 0, 0, 0};
  v8i g4 = {0, 0, 0, 0, 0, 0, 0, 0};
  __builtin_amdgcn_tensor_load_to_lds(g0, g1, g2, g3, g4, 0);
}

// ---------------------------------------------------------------------------
// WMMA f32 GEMM: Hs[row] = (X @ W)[row] * dinv[row]; Agg initialized = Hs.
// Block = 8 waves; wave w owns M-tile (blockIdx.x*8+w) and the full N strip.
// W staged once into LDS by the TDM; per k-step one global A fragment is
// reused across all N-tiles (B frags served from LDS).
//   A frag (16x4 f32, 2 VGPR): lanes 0-15 -> M=lane, K=k0..k0+1;
//                              lanes 16-31 -> M=lane-16, K=k0+2..k0+3
//   C/D (16x16 f32, 8 VGPR):   vgpr r -> M = r + 8*(lane>=16), N = lane&15
// EXEC stays all-ones: A rows are clamped (no predication), stores guarded.
// ---------------------------------------------------------------------------
template <int N_DIM, int K_DIM>
__global__ void gcn_gemm_wmma(const float* __restrict__ X,
                              const float* __restrict__ W,
                              const float* __restrict__ dinv,
                              float* __restrict__ Hs,
                              float* __restrict__ Agg,
                              int M) {
  constexpr int NT = N_DIM / 16;  // N tiles per wave
  __shared__ float sW[K_DIM * N_DIM];

  if (threadIdx.x == 0) {
    tdm_load_w(W, (unsigned)(uintptr_t)&sW[0], N_DIM, K_DIM);
    __builtin_amdgcn_s_wait_tensorcnt(0);
  }
  __syncthreads();

  const int lane = threadIdx.x & 31;
  const int wv   = threadIdx.x >> 5;
  const int half = lane >> 4;   // selects K pair
  const int l15  = lane & 15;

  const int tilesM = (M + 15) >> 4;
  int tileM = blockIdx.x * 8 + wv;
  if (tileM >= tilesM) tileM = tilesM - 1;   // duplicate tile: same values, benign

  int rowA = tileM * 16 + l15;
  if (rowA > M - 1) rowA = M - 1;            // branch-free clamp, EXEC stays full

  v8f c[NT];
#pragma unroll
  for (int nt = 0; nt < NT; ++nt) c[nt] = (v8f){};

  const float* __restrict__ xrow = X + (size_t)rowA * K_DIM;
#pragma unroll 4
  for (int k0 = 0; k0 < K_DIM; k0 += 4) {
    const int ka = k0 + half * 2;
    const float2 av = *(const float2*)(xrow + ka);
    v2f a; a.x = av.x; a.y = av.y;
#pragma unroll
    for (int nt = 0; nt < NT; ++nt) {
      v2f b;
      b.x = sW[ka * N_DIM + nt * 16 + l15];
      b.y = sW[(ka + 1) * N_DIM + nt * 16 + l15];
      c[nt] = __builtin_amdgcn_wmma_f32_16x16x4_f32(
          false, a, false, b, (short)0, c[nt], false, false);
    }
  }

#pragma unroll
  for (int r = 0; r < 8; ++r) {
    const int row = tileM * 16 + r + half * 8;
    if (row < M) {
      const float s = dinv[row];               // pre-scale by dinv[src]
      const size_t base = (size_t)row * N_DIM + l15;
#pragma unroll
      for (int nt = 0; nt < NT; ++nt) {
        const float v = c[nt][r] * s;
        Hs[base + nt * 16]  = v;
        Agg[base + nt * 16] = v;               // self-loop term seeds the sum
      }
    }
  }
}

// ---------------------------------------------------------------------------
// Edge aggregation: one wave per edge; each lane moves C/32 channels with a
// vector gather from Hs[src] and f32 atomic adds into Agg[dst]. All traffic
// is L2-resident (H is 25.6 MB << 192 MB L2).
// ---------------------------------------------------------------------------
template <int C>
__global__ void edge_aggregate_kernel(const int* __restrict__ src,
                                      const int* __restrict__ dst,
                                      const float* __restrict__ Hs,
                                      float* __restrict__ Agg, int E) {
  const int lane = threadIdx.x & 31;
  const int edge = (int)((blockIdx.x * (size_t)blockDim.x + threadIdx.x) >> 5);
  if (edge >= E) return;
  const int s = src[edge];
  const int d = dst[edge];
  constexpr int V = C / 32;  // floats per lane (4 for C=128, 2 for C=64)
  const float* p = Hs + (size_t)s * C + lane * V;
  float* q = Agg + (size_t)d * C + lane * V;
  if constexpr (V == 4) {
    const float4 v = *(const float4*)p;
    atomicAdd(q + 0, v.x);
    atomicAdd(q + 1, v.y);
    atomicAdd(q + 2, v.z);
    atomicAdd(q + 3, v.w);
  } else {
    const float2 v = *(const float2*)p;
    atomicAdd(q + 0, v.x);
    atomicAdd(q + 1, v.y);
  }
}

// ---------------------------------------------------------------------------
// out[i][c] = Agg[i][c] * dinv[i] + b[c]  (+ optional ReLU)
// ---------------------------------------------------------------------------
template <bool RELU>
__global__ void finalize_kernel(const float* __restrict__ Agg,
                                const float* __restrict__ dinv,
                                const float* __restrict__ bias,
                                float* __restrict__ out, int n, int C) {
  const int i = blockIdx.x * blockDim.x + threadIdx.x;
  if (i >= n * C) return;
  const int row = i / C;
  const int col = i - row * C;
  float v = Agg[i] * dinv[row] + bias[col];
  if (RELU) v = fmaxf(v, 0.0f);
  out[i] = v;
}

// ---------------------------------------------------------------------------
extern "C" void kernel_launch(void* const* d_in, const int* in_sizes, int n_in,
                              void* d_out, int out_size, void* d_ws, size_t ws_size,
                              hipStream_t stream) {
  const float* x   = (const float*)d_in[0];
  const int*   ei  = (const int*)d_in[1];   // [2, E] int32
  const float* W1  = (const float*)d_in[2];
  const float* b1  = (const float*)d_in[3];
  const float* W2  = (const float*)d_in[4];
  const float* b2  = (const float*)d_in[5];
  float*       out = (float*)d_out;

  const int N = in_sizes[0] / IN_C;   // 50000
  const int E = in_sizes[1] / 2;      // 800000
  const int* src = ei;
  const int* dst = ei + E;

  // Workspace layout (floats)
  const size_t nPad = ((size_t)N + 63) & ~(size_t)63;
  float* deg  = (float*)d_ws;
  float* dinv = deg  + nPad;
  float* Hs   = dinv + nPad;                 // N x 128 (reused N x 64 in layer 2)
  float* Agg  = Hs   + (size_t)N * HID_C;    // N x 128 (reused N x 64)
  float* H1   = Agg  + (size_t)N * HID_C;    // N x 128 hidden activations

  const int tilesM   = (N + 15) / 16;
  const int gemmBlks = (tilesM + 7) / 8;     // 8 M-tiles (waves) per block

  // --- normalization coefficients (shared by both layers) ---
  deg_init_kernel<<<(N + 255) / 256, 256, 0, stream>>>(deg, N);
  deg_accum_kernel<<<(E + 255) / 256, 256, 0, stream>>>(dst, deg, E);
  dinv_kernel<<<(N + 255) / 256, 256, 0, stream>>>(deg, dinv, N);

  // --- layer 1: Hs = (X @ W1) * dinv ; agg ; H1 = relu(agg*dinv + b1) ---
  gcn_gemm_wmma<HID_C, IN_C><<<gemmBlks, 256, 0, stream>>>(
      x, W1, dinv, Hs, Agg, N);
  {
    const size_t threads = (size_t)E * 32;
    edge_aggregate_kernel<HID_C><<<(unsigned)((threads + 255) / 256), 256, 0, stream>>>(
        src, dst, Hs, Agg, E);
  }
  finalize_kernel<true><<<((size_t)N * HID_C + 255) / 256, 256, 0, stream>>>(
      Agg, dinv, b1, H1, N, HID_C);

  // --- layer 2: Hs = (H1 @ W2) * dinv ; agg ; out = agg*dinv + b2 ---
  gcn_gemm_wmma<OUT_C, HID_C><<<gemmBlks, 256, 0, stream>>>(
      H1, W2, dinv, Hs, Agg, N);
  {
    const size_t threads = (size_t)E * 32;
    edge_aggregate_kernel<OUT_C><<<(unsigned)((threads + 255) / 256), 256, 0, stream>>>(
        src, dst, Hs, Agg, E);
  }
  finalize_kernel<false><<<((size_t)N * OUT_C + 255) / 256, 256, 0, stream>>>(
      Agg, dinv, b2, out, N, OUT_C);
}